// MultiSimilarityLoss_82755429859993
// MI455X (gfx1250) — compile-verified
//
#include <hip/hip_runtime.h>

#define NB     8192
#define DK     128
#define THRESH 0.5f
#define MARGIN 0.1f
#define SPOS   2.0f
#define SNEG   50.0f
#define EPSI   1e-5f

typedef __attribute__((ext_vector_type(2))) float v2f;
typedef __attribute__((ext_vector_type(4))) float v4f;
typedef __attribute__((ext_vector_type(8))) float v8f;

// ---------------------------------------------------------------------------
// Kernel 0: zero the scalar output (graph replays don't re-poison buffers)
// ---------------------------------------------------------------------------
__global__ void msl_zero(float* out) {
    if (threadIdx.x == 0) out[0] = 0.0f;
}

// ---------------------------------------------------------------------------
// Kernel 1: row L2-normalize + permute into WMMA-operand-friendly layout.
// Permuted row layout: xnp[row][h][t][b] = xn[row][4t + 2h + b],  h in {0,1},
// t in 0..31, b in {0,1}.  A lane of the 16x4-f32 WMMA operand (lane&15 = row,
// lane>>4 = h) then reads the first 32 floats (t < 16) of its K-stream CONTIGUOUSLY ->
// 8 x global_load_b128 per tile instead of 16 x b64.
// ---------------------------------------------------------------------------
__global__ void msl_normalize(const float* __restrict__ feats,
                              float* __restrict__ xnp) {
    const int wave = threadIdx.x >> 5;
    const int lane = threadIdx.x & 31;
    const int row  = blockIdx.x * 8 + wave;

    float4 v = reinterpret_cast<const float4*>(feats + row * DK)[lane];
    float ss = v.x * v.x + v.y * v.y + v.z * v.z + v.w * v.w;
#pragma unroll
    for (int m = 16; m >= 1; m >>= 1) ss += __shfl_xor(ss, m, 32);
    float inv = 1.0f / fmaxf(sqrtf(ss), 1e-12f);
    v.x *= inv; v.y *= inv; v.z *= inv; v.w *= inv;

    // k = 4*lane + {0,1} -> (h=0, t=lane); k = 4*lane + {2,3} -> (h=1, t=lane)
    float* base = xnp + row * DK;
    v2f lo; lo[0] = v.x; lo[1] = v.y;
    v2f hi; hi[0] = v.z; hi[1] = v.w;
    *reinterpret_cast<v2f*>(base + lane * 2)      = lo;
    *reinterpret_cast<v2f*>(base + 64 + lane * 2) = hi;
}

// ---------------------------------------------------------------------------
// Load one 16-row x 64-K operand tile as 8 x b128 into independent VGPRs.
// base pointer is 256B aligned (row*128 + half*64 floats).
// ---------------------------------------------------------------------------
__device__ __forceinline__ void load_tile(const float* __restrict__ xnp,
                                          int base_row, int lane, v4f (&b)[8]) {
    const float* p = xnp + (base_row + (lane & 15)) * DK + ((lane >> 4) << 6);
#pragma unroll
    for (int j = 0; j < 8; ++j)
        b[j] = reinterpret_cast<const v4f*>(p)[j];
}

// 16 WMMAs split over TWO independent accumulator chains for XDL-pipe ILP;
// tile j holds K-chunks t=2j (lo half of the b128) and t=2j+1 (hi half).
__device__ __forceinline__ v8f tile_gemm(const v4f (&a)[8], const v4f (&b)[8]) {
    v8f c0 = {}, c1 = {};
#pragma unroll
    for (int j = 0; j < 8; ++j) {
        v2f alo = __builtin_shufflevector(a[j], a[j], 0, 1);
        v2f blo = __builtin_shufflevector(b[j], b[j], 0, 1);
        c0 = __builtin_amdgcn_wmma_f32_16x16x4_f32(false, alo, false, blo,
                                                   (short)0, c0, false, false);
        v2f ahi = __builtin_shufflevector(a[j], a[j], 2, 3);
        v2f bhi = __builtin_shufflevector(b[j], b[j], 2, 3);
        c1 = __builtin_amdgcn_wmma_f32_16x16x4_f32(false, ahi, false, bhi,
                                                   (short)0, c1, false, false);
    }
    return c0 + c1;
}

// ---------------------------------------------------------------------------
// Kernel 2 (pass 1): per-row max over negatives, min over valid positives.
// block = 8 waves owns 16 rows; wave w covers column tiles w, w+8, ... (64),
// double-buffered so next tile's 8 b128 loads overlap current tile's WMMAs.
// ---------------------------------------------------------------------------
__global__ void msl_pass1(const float* __restrict__ xnp,
                          const int* __restrict__ labels,
                          float* __restrict__ mneg,
                          float* __restrict__ mpos) {
    __shared__ float s_max[8][16];
    __shared__ float s_min[8][16];
    const int lane    = threadIdx.x & 31;
    const int wave    = threadIdx.x >> 5;
    const int rowbase = blockIdx.x * 16;
    const int half    = lane >> 4;

    v4f a[8];
    load_tile(xnp, rowbase, lane, a);

    int lab_i[8];
#pragma unroll
    for (int r = 0; r < 8; ++r) lab_i[r] = labels[rowbase + r + 8 * half];

    float mx[8], mn[8];
#pragma unroll
    for (int r = 0; r < 8; ++r) { mx[r] = -__builtin_inff(); mn[r] = __builtin_inff(); }

    auto epilogue = [&](const v8f& c, int colbase) {
        const int lab_j = labels[colbase + (lane & 15)];
#pragma unroll
        for (int r = 0; r < 8; ++r) {
            const float sim  = c[r];
            const bool  same = (lab_i[r] == lab_j);
            if (!same)                     mx[r] = fmaxf(mx[r], sim);
            if (same && sim < 1.0f - EPSI) mn[r] = fminf(mn[r], sim);
        }
    };

    v4f b0[8], b1[8];
    int ct = wave;                        // exactly 64 tiles per wave, uniform
    load_tile(xnp, ct * 16, lane, b0);
#pragma unroll 1
    for (int it = 0; it < 64; it += 2) {
        load_tile(xnp, (ct + 8) * 16, lane, b1);
        epilogue(tile_gemm(a, b0), ct * 16);
        if (it + 2 < 64) load_tile(xnp, (ct + 16) * 16, lane, b0);
        epilogue(tile_gemm(a, b1), (ct + 8) * 16);
        ct += 16;
    }

    // reduce across the 16 lanes of each C-matrix half (masks stay in-half)
#pragma unroll
    for (int r = 0; r < 8; ++r) {
#pragma unroll
        for (int m = 1; m <= 8; m <<= 1) {
            mx[r] = fmaxf(mx[r], __shfl_xor(mx[r], m, 32));
            mn[r] = fminf(mn[r], __shfl_xor(mn[r], m, 32));
        }
    }
    if ((lane & 15) == 0) {
#pragma unroll
        for (int r = 0; r < 8; ++r) {
            s_max[wave][r + 8 * half] = mx[r];
            s_min[wave][r + 8 * half] = mn[r];
        }
    }
    __syncthreads();
    if (threadIdx.x < 16) {
        float vmax = -__builtin_inff(), vmin = __builtin_inff();
#pragma unroll
        for (int w = 0; w < 8; ++w) {
            vmax = fmaxf(vmax, s_max[w][threadIdx.x]);
            vmin = fminf(vmin, s_min[w][threadIdx.x]);
        }
        mneg[rowbase + threadIdx.x] = vmax;   // -inf  <=> no negative exists
        mpos[rowbase + threadIdx.x] = vmin;   // +inf  <=> no positive exists
    }
}

// ---------------------------------------------------------------------------
// Kernel 3 (pass 2): recompute sim tiles, hard-mine, accumulate exp-sums,
// per-row log1p, block partial -> atomicAdd into out (pre-divided by B).
// ---------------------------------------------------------------------------
__global__ void msl_pass2(const float* __restrict__ xnp,
                          const int* __restrict__ labels,
                          const float* __restrict__ mneg,
                          const float* __restrict__ mpos,
                          float* __restrict__ out) {
    __shared__ float s_pos[8][16];
    __shared__ float s_neg[8][16];
    const int lane    = threadIdx.x & 31;
    const int wave    = threadIdx.x >> 5;
    const int rowbase = blockIdx.x * 16;
    const int half    = lane >> 4;

    v4f a[8];
    load_tile(xnp, rowbase, lane, a);

    int   lab_i[8];
    float thr_n[8], thr_p[8];
    bool  has_n[8], has_p[8];
#pragma unroll
    for (int r = 0; r < 8; ++r) {
        const int i = rowbase + r + 8 * half;
        lab_i[r] = labels[i];
        thr_n[r] = mneg[i];
        thr_p[r] = mpos[i];
        has_n[r] = (thr_n[r] > -2.0f);   // cosine sims live in [-1, 1]
        has_p[r] = (thr_p[r] <  2.0f);
    }

    float psum[8], nsum[8];
#pragma unroll
    for (int r = 0; r < 8; ++r) { psum[r] = 0.0f; nsum[r] = 0.0f; }

    auto epilogue = [&](const v8f& c, int colbase) {
        const int lab_j = labels[colbase + (lane & 15)];
#pragma unroll
        for (int r = 0; r < 8; ++r) {
            const float sim  = c[r];
            const bool  same = (lab_i[r] == lab_j);
            const bool  pos  = same && (sim < 1.0f - EPSI);
            if (pos && has_n[r] && (sim - MARGIN < thr_n[r]))
                psum[r] += __expf(-SPOS * (sim - THRESH));
            if (!same && has_p[r] && (sim + MARGIN > thr_p[r]))
                nsum[r] += __expf(SNEG * (sim - THRESH));
        }
    };

    v4f b0[8], b1[8];
    int ct = wave;
    load_tile(xnp, ct * 16, lane, b0);
#pragma unroll 1
    for (int it = 0; it < 64; it += 2) {
        load_tile(xnp, (ct + 8) * 16, lane, b1);
        epilogue(tile_gemm(a, b0), ct * 16);
        if (it + 2 < 64) load_tile(xnp, (ct + 16) * 16, lane, b0);
        epilogue(tile_gemm(a, b1), (ct + 8) * 16);
        ct += 16;
    }

#pragma unroll
    for (int r = 0; r < 8; ++r) {
#pragma unroll
        for (int m = 1; m <= 8; m <<= 1) {
            psum[r] += __shfl_xor(psum[r], m, 32);
            nsum[r] += __shfl_xor(nsum[r], m, 32);
        }
    }
    if ((lane & 15) == 0) {
#pragma unroll
        for (int r = 0; r < 8; ++r) {
            s_pos[wave][r + 8 * half] = psum[r];
            s_neg[wave][r + 8 * half] = nsum[r];
        }
    }
    __syncthreads();
    if (threadIdx.x < 16) {
        float ps = 0.0f, ns = 0.0f;
#pragma unroll
        for (int w = 0; w < 8; ++w) { ps += s_pos[w][threadIdx.x]; ns += s_neg[w][threadIdx.x]; }
        float loss = 0.0f;
        if (ps > 0.0f) loss += log1pf(ps) * (1.0f / SPOS);   // any(pos_sel) <=> ps>0
        if (ns > 0.0f) loss += log1pf(ns) * (1.0f / SNEG);
        // 16 row losses live in lanes 0..15 of wave 0 -> in-half tree sum
#pragma unroll
        for (int m = 1; m <= 8; m <<= 1) loss += __shfl_xor(loss, m, 32);
        if (threadIdx.x == 0) atomicAdd(out, loss * (1.0f / (float)NB));
    }
}

// ---------------------------------------------------------------------------
extern "C" void kernel_launch(void* const* d_in, const int* in_sizes, int n_in,
                              void* d_out, int out_size, void* d_ws, size_t ws_size,
                              hipStream_t stream) {
    const float* feats  = (const float*)d_in[0];
    const int*   labels = (const int*)d_in[1];
    float*       out    = (float*)d_out;

    float* xnp  = (float*)d_ws;          // 8192*128 f32 = 4 MB (fits in L2)
    float* mneg = xnp + (size_t)NB * DK; // 32 KB
    float* mpos = mneg + NB;             // 32 KB

    msl_zero     <<<1,     32, 0, stream>>>(out);
    msl_normalize<<<NB/8, 256, 0, stream>>>(feats, xnp);
    msl_pass1    <<<NB/16, 256, 0, stream>>>(xnp, labels, mneg, mpos);
    msl_pass2    <<<NB/16, 256, 0, stream>>>(xnp, labels, mneg, mpos, out);
}